// TopK_13434657702726
// MI455X (gfx1250) — compile-verified
//
#include <hip/hip_runtime.h>

#define NB   128   // batch B
#define NN   1024  // N (top-k reduction axis)
#define NM   1024  // M (weight rows)
#define TOPK 128

typedef __attribute__((ext_vector_type(2))) float v2f;
typedef __attribute__((ext_vector_type(8))) float v8f;

// Order-preserving float -> uint32 key (larger key == larger float)
__device__ __forceinline__ uint32_t f2key(float f) {
    uint32_t u = __float_as_uint(f);
    return (u & 0x80000000u) ? ~u : (u | 0x80000000u);
}
__device__ __forceinline__ float key2f(uint32_t k) {
    uint32_t u = (k & 0x80000000u) ? (k & 0x7FFFFFFFu) : ~k;
    return __uint_as_float(u);
}

// Full wave32 f32 sum, result broadcast to ALL lanes, using two chained
// v_wmma_f32_16x16x4_f32 with B = ones (no LDS / no ds_bpermute):
//   WMMA-1: D[i][j] = s(lane i) + s(lane i+16)            (B = ones)
//   sum of 8 D regs: lanes 0-15 -> sum(rows 0-7), lanes 16-31 -> sum(rows 8-15)
//   WMMA-2: each row-sum = low-half + high-half = total, broadcast everywhere.
// Exact for our use: counts are small ints, value sums are ordinary f32 adds (RNE).
__device__ __forceinline__ float wave_sum_f32(float s) {
    v2f bo; bo[0] = 1.0f; bo[1] = 1.0f;
    v8f cz = {0.f, 0.f, 0.f, 0.f, 0.f, 0.f, 0.f, 0.f};

    v2f a1; a1[0] = s; a1[1] = 0.0f;
    v8f d1 = __builtin_amdgcn_wmma_f32_16x16x4_f32(
        false, a1, false, bo, (short)0, cz, false, false);
    float t = ((d1[0] + d1[1]) + (d1[2] + d1[3])) + ((d1[4] + d1[5]) + (d1[6] + d1[7]));

    v2f a2; a2[0] = t; a2[1] = 0.0f;
    v8f d2 = __builtin_amdgcn_wmma_f32_16x16x4_f32(
        false, a2, false, bo, (short)0, cz, false, false);
    return d2[0];   // identical in every lane
}

// One wave32 per (b, m) pair; each wave processes 4 consecutive m.
// Lane holds 32 of the 1024 elements as 8 float4s (n = i*128 + lane*4 + j).
__global__ __launch_bounds__(256) void topk_sum_relu_kernel(
    const float* __restrict__ x, const float* __restrict__ w,
    const float* __restrict__ bias, const float* __restrict__ Tp,
    float* __restrict__ out)
{
    const int lane = threadIdx.x & 31;
    const int wid  = threadIdx.x >> 5;
    const int b    = blockIdx.x >> 5;                  // 32 blocks per batch row
    const int mb   = (blockIdx.x & 31) * 32 + wid * 4; // 8 waves * 4 m = 32 m/block
    const float Tval = Tp[0];

    // Cache x[b,:] in registers; reused across 4 weight rows (coalesced float4)
    const float4* xr = reinterpret_cast<const float4*>(x + (size_t)b * NN);
    float4 xv[8];
    #pragma unroll
    for (int i = 0; i < 8; ++i) xv[i] = xr[i * 32 + lane];

    for (int j = 0; j < 4; ++j) {
        const int m = mb + j;
        const float4* wr = reinterpret_cast<const float4*>(w + (size_t)m * NN);
        if (j + 1 < 4)  // warm caches for next weight row (global_prefetch_b8)
            __builtin_prefetch(w + (size_t)(m + 1) * NN + lane * 4, 0, 3);

        // Sortable keys of v = x + w, fully register-resident (32 per lane)
        uint32_t kk[32];
        #pragma unroll
        for (int i = 0; i < 8; ++i) {
            float4 wv = wr[i * 32 + lane];
            kk[i * 4 + 0] = f2key(xv[i].x + wv.x);
            kk[i * 4 + 1] = f2key(xv[i].y + wv.y);
            kk[i * 4 + 2] = f2key(xv[i].z + wv.z);
            kk[i * 4 + 3] = f2key(xv[i].w + wv.w);
        }

        // Bitwise radix-select of the exact 128th-largest key.
        // remf = rank still needed among keys matching 'prefix' on decided bits.
        // (kk & ~(2^bit - 1)) == cand  <=>  (kk - cand) < 2^bit  (unsigned; cand's
        // low bits are zero so the interval never wraps).
        uint32_t prefix = 0u;
        float remf = (float)TOPK;        // counts are small ints: exact in f32
        for (int bit = 31; bit >= 0; --bit) {
            const uint32_t cand = prefix | (1u << bit);
            const uint32_t span = 1u << bit;
            uint32_t c = 0u;
            #pragma unroll
            for (int e = 0; e < 32; ++e) c += ((kk[e] - cand) < span) ? 1u : 0u;
            const float ctot = wave_sum_f32((float)c);  // broadcast, matrix pipe
            if (ctot >= remf) prefix = cand;
            else              remf -= ctot;
        }
        // #(key > prefix) == TOPK - remf; include 'remf' ties at value(prefix).

        float s = 0.0f;
        #pragma unroll
        for (int e = 0; e < 32; ++e)
            s += (kk[e] > prefix) ? key2f(kk[e]) : 0.0f;

        float tot = wave_sum_f32(s) + remf * key2f(prefix);

        if (lane == 0)
            out[(size_t)b * NM + m] = bias[m] + fmaxf(tot - Tval, 0.0f);
    }
}

extern "C" void kernel_launch(void* const* d_in, const int* in_sizes, int n_in,
                              void* d_out, int out_size, void* d_ws, size_t ws_size,
                              hipStream_t stream) {
    const float* x    = (const float*)d_in[0];  // (128, 1024)
    const float* w    = (const float*)d_in[1];  // (1024, 1024)
    const float* bias = (const float*)d_in[2];  // (1024,)
    const float* T    = (const float*)d_in[3];  // (1,)
    float* out = (float*)d_out;                 // (128, 1024) float32

    dim3 grid(NB * (NM / 32));   // 4096 blocks
    dim3 block(256);             // 8 wave32 per block
    hipLaunchKernelGGL(topk_sum_relu_kernel, grid, block, 0, stream,
                       x, w, bias, T, out);
}